// PosEncodedScheduler_49898930045516
// MI455X (gfx1250) — compile-verified
//
#include <hip/hip_runtime.h>
#include <hip/hip_bf16.h>
#include <math.h>

#define MODEL_DIM 128
#define POS_DIM   33
#define N_FRAMES  128
#define N_SAMPLES 32768
#define N_ROWS    512          // B(32) * N_EVENTS(16)
#define PI_F      3.14159265358979323846f

typedef __attribute__((ext_vector_type(2))) float v2f;
typedef __attribute__((ext_vector_type(4))) float v4f;
typedef __attribute__((ext_vector_type(8))) float v8f;

// ---------------------------------------------------------------------------
// Kernel 1: per-row frame index via WMMA fp32 matmul + argmax.
// 32 blocks, 1 wave (32 threads) each; block `tile` owns rows tile*16..+15.
// ---------------------------------------------------------------------------
__global__ __launch_bounds__(32) void sched_argmax_kernel(
    const float* __restrict__ TL,     // (512, 128)
    const float* __restrict__ W,      // (128, 33)
    const float* __restrict__ bias,   // (33,)
    int* __restrict__ sidx)           // (512,) sample shift per row
{
    __shared__ float posn[N_FRAMES * POS_DIM];  // normalized pos encoding
    __shared__ float Pm[16 * 36];               // P tile, padded row stride

    const int l    = threadIdx.x;
    const int tile = blockIdx.x;

    // --- build normalized positional-encoding table in LDS ---
    for (int f = l; f < N_FRAMES; f += 32) {
        float x = -PI_F + (2.0f * PI_F / 127.0f) * (float)f;
        float e[POS_DIM];
        e[0] = x;
        float ss = x * x;
        float sc = 1.0f;
        #pragma unroll
        for (int i = 0; i < 16; ++i) {
            float xi = x * sc;
            float sv = sinf(xi);
            float cv = cosf(xi);
            e[1 + 2 * i] = sv;
            e[2 + 2 * i] = cv;
            ss += sv * sv + cv * cv;
            sc *= 2.0f;
        }
        float inv = 1.0f / (sqrtf(ss) + 1e-8f);
        for (int j = 0; j < POS_DIM; ++j)
            posn[f * POS_DIM + j] = e[j] * inv;
    }

    // --- P = TL_tile(16x128) @ W(128x33->pad48) via V_WMMA_F32_16X16X4_F32 ---
    // A 16x4 layout: lane l -> M = l&15; VGPR0/1 hold K = koff, koff+1 with
    // koff = (l>>4)*2.  B 4x16: N = l&15, K rows mirror A.  C/D: 8 VGPRs,
    // M = v + 8*(l>>4), N = l&15.
    const int mrow = l & 15;
    const int koff = (l >> 4) * 2;
    const int ncol = l & 15;
    const bool is_c32 = (ncol == 0);   // lane that owns padded column 32

    v8f acc0 = {}, acc1 = {}, acc2 = {};
    const float* Arow = TL + (size_t)(tile * 16 + mrow) * MODEL_DIM;

    for (int k = 0; k < MODEL_DIM; k += 4) {
        v2f a;
        a.x = Arow[k + koff];
        a.y = Arow[k + koff + 1];

        const float* Wr0 = W + (size_t)(k + koff) * POS_DIM;
        const float* Wr1 = Wr0 + POS_DIM;

        v2f b0, b1, b2;
        b0.x = Wr0[ncol];            b0.y = Wr1[ncol];
        b1.x = Wr0[16 + ncol];       b1.y = Wr1[16 + ncol];
        // Column 32 is a valid address for every lane: load unconditionally
        // (uniform) and select -> v_cndmask, no EXEC-divergent branches.
        float w32a = Wr0[32];
        float w32b = Wr1[32];
        b2.x = is_c32 ? w32a : 0.0f;
        b2.y = is_c32 ? w32b : 0.0f;

        acc0 = __builtin_amdgcn_wmma_f32_16x16x4_f32(false, a, false, b0, (short)0, acc0, false, false);
        acc1 = __builtin_amdgcn_wmma_f32_16x16x4_f32(false, a, false, b1, (short)0, acc1, false, false);
        acc2 = __builtin_amdgcn_wmma_f32_16x16x4_f32(false, a, false, b2, (short)0, acc2, false, false);
    }

    // --- add bias, spill P tile (only the 33 real columns) to LDS ---
    const int mbase = (l >> 4) * 8;
    {
        float bv0 = bias[ncol];
        float bv1 = bias[16 + ncol];
        #pragma unroll
        for (int v = 0; v < 8; ++v) {
            Pm[(mbase + v) * 36 + ncol]      = acc0[v] + bv0;
            Pm[(mbase + v) * 36 + 16 + ncol] = acc1[v] + bv1;
        }
        if (is_c32) {   // column 32
            float bv2 = bias[32];
            #pragma unroll
            for (int v = 0; v < 8; ++v)
                Pm[(mbase + v) * 36 + 32] = acc2[v] + bv2;
        }
    }
    __syncthreads();

    // --- argmax over 128 frames: half-wave splits the frame range ---
    const int row   = l & 15;
    const int fbase = (l >> 4) * 64;
    float bestv = -3.0e38f;
    int   besti = 0;
    for (int f = fbase; f < fbase + 64; ++f) {
        float dot = 0.0f;
        #pragma unroll
        for (int j = 0; j < POS_DIM; ++j)
            dot += Pm[row * 36 + j] * posn[f * POS_DIM + j];
        if (dot > bestv) { bestv = dot; besti = f; }
    }
    float ov = __shfl_xor(bestv, 16);
    int   oi = __shfl_xor(besti, 16);
    if (ov > bestv || (ov == bestv && oi < besti)) { bestv = ov; besti = oi; }
    if (l < 16)
        sidx[tile * 16 + row] = besti * 256;   // first-occurrence argmax * step
}

// ---------------------------------------------------------------------------
// Kernel 2: bandwidth kernel.  out[r,t] = t>=s ? stems[r,t-s]/256 : 0.
// s is a multiple of 256 -> float4 alignment preserved under the shift.
// 128 MB total HBM traffic, zero reuse -> non-temporal B128 streaming.
// ---------------------------------------------------------------------------
__global__ __launch_bounds__(256) void shift_copy_kernel(
    const float* __restrict__ stems,  // (512, 32768)
    const int* __restrict__ sidx,     // (512,)
    float* __restrict__ out)          // (512, 32768)
{
    const int row = blockIdx.y;
    const int t4  = blockIdx.x * blockDim.x + threadIdx.x;   // 0..8191
    const int t   = t4 << 2;
    const int s   = sidx[row];

    const v4f* __restrict__ src =
        reinterpret_cast<const v4f*>(stems + (size_t)row * N_SAMPLES);
    v4f* __restrict__ dst =
        reinterpret_cast<v4f*>(out + (size_t)row * N_SAMPLES);

    v4f v = {0.0f, 0.0f, 0.0f, 0.0f};
    if (t >= s) {
        const float kScale = 1.0f / 256.0f;   // 1/sqrt(2*N_SAMPLES), ortho FFT pair
        v4f r = __builtin_nontemporal_load(&src[(t - s) >> 2]);
        v = r * kScale;
    }
    __builtin_nontemporal_store(v, &dst[t4]);
}

// ---------------------------------------------------------------------------
extern "C" void kernel_launch(void* const* d_in, const int* in_sizes, int n_in,
                              void* d_out, int out_size, void* d_ws, size_t ws_size,
                              hipStream_t stream) {
    const float* time_latent = (const float*)d_in[0];  // (512, 128)
    const float* stems       = (const float*)d_in[1];  // (32, 16, 32768)
    // d_in[2] = targets: unused by the reference computation
    const float* W           = (const float*)d_in[3];  // (128, 33)
    const float* bias        = (const float*)d_in[4];  // (33,)

    float* out  = (float*)d_out;
    int*   sidx = (int*)d_ws;                          // 512 ints of scratch

    // Kernel 1: 32 tiles of 16 rows, one wave32 each (WMMA f32 path).
    sched_argmax_kernel<<<dim3(N_ROWS / 16), dim3(32), 0, stream>>>(
        time_latent, W, bias, sidx);

    // Kernel 2: 512 rows x 8192 float4, 256 threads/block.
    dim3 grid2(N_SAMPLES / 4 / 256, N_ROWS);
    shift_copy_kernel<<<grid2, dim3(256), 0, stream>>>(stems, sidx, out);
}